// DNN_SKalmanNet_21492016349239
// MI455X (gfx1250) — compile-verified
//
#include <hip/hip_runtime.h>
#include <cstdint>

// ---------------------------------------------------------------------------
// DNN_SKalmanNet on MI455X (gfx1250): fp32 GEMV chain, HBM-bandwidth-bound.
// GEMV mapped onto V_WMMA_F32_16X16X4_F32:
//   D = A x B, A[m][k] = x[k0+k] (broadcast over m), B[k][n] = W[tile*16+n][k0+k]
//   => D[m][n] = sum_k W[row_n][k]*x[k]  (same for every m); accumulate over k0.
// 4 waves per block split the K range; deterministic LDS reduction.
// Inner loop uses pointer-bumping so loads get immediate offsets (no per-trip
// 64-bit index math), 4 WMMAs per trip.
// ---------------------------------------------------------------------------

typedef __attribute__((ext_vector_type(2))) float v2f;
typedef __attribute__((ext_vector_type(8))) float v8f;

#define ACT_NONE 0
#define ACT_RELU 1

// Model dims
#define XD   32
#define YD   32
#define IN2  128        // 2*XD + 2*YD
#define H1D  1344       // (XD+YD)*3*7
#define H2D  4096       // XD*YD*4
#define GHD  4096       // 2*(XD*XD + YD*YD)

__global__ __launch_bounds__(128) void concat4_kernel(
    const float* __restrict__ a, const float* __restrict__ b,
    const float* __restrict__ c, const float* __restrict__ d,
    float* __restrict__ x)
{
    const int i = threadIdx.x;           // 0..127
    float v;
    if (i < 32)        v = a[i];
    else if (i < 64)   v = b[i - 32];
    else if (i < 96)   v = c[i - 64];
    else               v = d[i - 96];
    x[i] = v;
}

// y[tile*16 .. tile*16+15] = act(W[rows, K] @ x + bias)
// Grid.x = M/16 (M a multiple of 16), block = 128 threads (4 waves).
// K must be a multiple of 64 (each wave's K/4 segment is a multiple of 16).
__global__ __launch_bounds__(128) void gemv_wmma_kernel(
    const float* __restrict__ W, const float* __restrict__ x,
    const float* __restrict__ bias, float* __restrict__ y,
    int K, int act)
{
    const int tile = blockIdx.x;
    const int lane = threadIdx.x & 31;
    const int wave = threadIdx.x >> 5;
    const int half = lane >> 4;          // 0: lanes 0-15 (K=0,1), 1: lanes 16-31 (K=2,3)
    const int n    = lane & 15;          // output row index within the 16-row tile

    const int kseg  = K >> 2;            // per-wave K segment (multiple of 16)
    const int iters = kseg >> 4;         // 16 K-values per trip

    // Lane-skewed streaming pointers (skew 2*half hoisted out of the loop).
    const float* __restrict__ wp =
        W + (size_t)(tile * 16 + n) * (size_t)K + (size_t)(wave * kseg + 2 * half);
    const float* __restrict__ xp = x + (wave * kseg + 2 * half);

    v8f acc = {0.f, 0.f, 0.f, 0.f, 0.f, 0.f, 0.f, 0.f};

    for (int it = 0; it < iters; ++it) {
        // stream-ahead prefetch of this lane's W row (~2KB ahead)
        __builtin_prefetch(wp + 512, 0, 0);
#pragma unroll
        for (int kk = 0; kk < 16; kk += 4) {
            // A (16x4 f32): VGPR0 = {K0 | K2}, VGPR1 = {K1 | K3} per lane half
            v2f av = *(const v2f*)(xp + kk);
            // B (4x16 f32): VGPR0 = {K0 | K2}, VGPR1 = {K1 | K3}, N = lane&15
            v2f bv = *(const v2f*)(wp + kk);
            // (neg_a, A, neg_b, B, c_mod, C, reuse_a, reuse_b)
            acc = __builtin_amdgcn_wmma_f32_16x16x4_f32(
                false, av, false, bv, (short)0, acc, false, false);
        }
        wp += 16;
        xp += 16;
    }

    // D layout: VGPR0, lanes 0-15 = D[M=0][N=lane] -> y partial for row tile*16+lane
    __shared__ float red[4][16];
    if (lane < 16) red[wave][n] = acc[0];
    __syncthreads();

    if (threadIdx.x < 16) {
        const int row = tile * 16 + threadIdx.x;
        float v = red[0][threadIdx.x] + red[1][threadIdx.x] +
                  red[2][threadIdx.x] + red[3][threadIdx.x] + bias[row];
        if (act == ACT_RELU) v = fmaxf(v, 0.f);
        y[row] = v;
    }
}

__global__ __launch_bounds__(256) void gru_combine_kernel(
    const float* __restrict__ gi, const float* __restrict__ gh,
    const float* __restrict__ h0, float* __restrict__ h)
{
    const int i = blockIdx.x * blockDim.x + threadIdx.x;
    if (i < GHD) {
        const float r = 1.f / (1.f + __expf(-(gi[i] + gh[i])));
        const float z = 1.f / (1.f + __expf(-(gi[GHD + i] + gh[GHD + i])));
        const float nn = tanhf(gi[2 * GHD + i] + r * gh[2 * GHD + i]);
        h[i] = (1.f - z) * nn + z * h0[i];
    }
}

extern "C" void kernel_launch(void* const* d_in, const int* in_sizes, int n_in,
                              void* d_out, int out_size, void* d_ws, size_t ws_size,
                              hipStream_t stream)
{
    (void)in_sizes; (void)n_in; (void)out_size; (void)ws_size;

    const float* state_inno = (const float*)d_in[0];
    const float* precov     = (const float*)d_in[1];
    const float* residual   = (const float*)d_in[2];
    const float* meas_cov   = (const float*)d_in[3];
    const float* W1   = (const float*)d_in[4];
    const float* b1   = (const float*)d_in[5];
    const float* w_ih = (const float*)d_in[6];
    const float* w_hh = (const float*)d_in[7];
    const float* b_ih = (const float*)d_in[8];
    const float* b_hh = (const float*)d_in[9];
    const float* h0   = (const float*)d_in[10];
    const float* W2a  = (const float*)d_in[11];
    const float* b2a  = (const float*)d_in[12];
    const float* W2b  = (const float*)d_in[13];
    const float* b2b  = (const float*)d_in[14];
    const float* W3a  = (const float*)d_in[15];
    const float* b3a  = (const float*)d_in[16];
    const float* W3b  = (const float*)d_in[17];
    const float* b3b  = (const float*)d_in[18];

    float* ws   = (float*)d_ws;
    float* xbuf = ws;                 // 128
    float* l1   = ws + 128;           // 1344
    float* gi   = ws + 1472;          // 12288
    float* gh   = ws + 13760;         // 12288
    float* h    = ws + 26048;         // 4096
    float* a2   = ws + 30144;         // 4096
    float* a3   = ws + 34240;         // 4096  (total 38336 floats = 150 KB)
    float* out  = (float*)d_out;      // [x_hat(32) | P_hat(32)]

    // 1) x = concat(state_inno, precov, residual, meas_cov)
    concat4_kernel<<<1, 128, 0, stream>>>(state_inno, precov, residual, meas_cov, xbuf);

    // 2) l1 = relu(W1 @ x + b1)            [1344 x 128]
    gemv_wmma_kernel<<<H1D / 16, 128, 0, stream>>>(W1, xbuf, b1, l1, IN2, ACT_RELU);

    // 3) gi = w_ih @ l1 + b_ih             [12288 x 1344]
    gemv_wmma_kernel<<<(3 * GHD) / 16, 128, 0, stream>>>(w_ih, l1, b_ih, gi, H1D, ACT_NONE);

    // 4) gh = w_hh @ h0 + b_hh             [12288 x 4096]  (201 MB — the big one)
    gemv_wmma_kernel<<<(3 * GHD) / 16, 128, 0, stream>>>(w_hh, h0, b_hh, gh, GHD, ACT_NONE);

    // 5) GRU gate combine -> h
    gru_combine_kernel<<<GHD / 256, 256, 0, stream>>>(gi, gh, h0, h);

    // 6) a2 = relu(W2a @ h + b2a); a3 = relu(W3a @ h + b3a)   [4096 x 4096] each
    gemv_wmma_kernel<<<H2D / 16, 128, 0, stream>>>(W2a, h, b2a, a2, GHD, ACT_RELU);
    gemv_wmma_kernel<<<H2D / 16, 128, 0, stream>>>(W3a, h, b3a, a3, GHD, ACT_RELU);

    // 7) x_hat = W2b @ a2 + b2b; P_hat = W3b @ a3 + b3b      [32 x 4096] each
    gemv_wmma_kernel<<<XD / 16, 128, 0, stream>>>(W2b, a2, b2b, out,      H2D, ACT_NONE);
    gemv_wmma_kernel<<<XD / 16, 128, 0, stream>>>(W3b, a3, b3b, out + XD, H2D, ACT_NONE);
}